// L2FunctionNorm_50173807952918
// MI455X (gfx1250) — compile-verified
//
#include <hip/hip_runtime.h>
#include <hip/hip_bf16.h>

typedef float v2f __attribute__((ext_vector_type(2)));
typedef float v8f __attribute__((ext_vector_type(8)));

#define WAVES_PER_BLOCK 8

// One wave per atom. D in {16, 48}, C == 32.
// t = S @ y  via V_WMMA_F32_16X16X4_F32, then per-channel bilinear norm,
// then y / (sqrt(norm) + eps) written back coalesced.
template <int D>
__global__ __launch_bounds__(256) void l2norm_wmma_kernel(
    const float* __restrict__ x,   // full [n_total, 32] input
    const float* __restrict__ S,   // [D, D] row-major
    float* __restrict__ out,       // full [n_total, 32] output
    int n_atoms,
    long long row_base)            // first basis row of this atom type
{
    constexpr int C  = 32;
    constexpr int MT = D / 16;     // 16-row output tiles
    constexpr int KC = D / 4;      // K=4 chunks
    constexpr int NV4 = D * C / 4; // float4 count of one atom's y

    __shared__ float s_S[D * D];
    __shared__ float s_y[WAVES_PER_BLOCK][D * C];

    const int tid  = threadIdx.x;
    const int wave = tid >> 5;
    const int lane = tid & 31;
    const int lo   = lane & 15;    // column / M index within 16
    const int hi   = lane >> 4;    // lane-half selector

    // Stage S into LDS once per block (read-only afterwards).
    for (int i = tid; i < D * D; i += 256) s_S[i] = S[i];
    __syncthreads();

    float* yl = &s_y[wave][0];
    float4* yl4 = (float4*)yl;

    const int stride = gridDim.x * WAVES_PER_BLOCK;
    for (int atom = blockIdx.x * WAVES_PER_BLOCK + wave; atom < n_atoms;
         atom += stride) {
        const long long rbase = (row_base + (long long)atom * D) * C;
        const float4* ya4 = (const float4*)(x + rbase);
        float4* oa4 = (float4*)(out + rbase);

        // ---- stage y[atom] into this wave's LDS slice (coalesced b128) ----
#pragma unroll
        for (int i = lane; i < NV4; i += 32) yl4[i] = ya4[i];
        // cross-lane LDS visibility within the wave: drain ds stores
        asm volatile("s_wait_dscnt 0" ::: "memory");

        // ---- t = S @ y via WMMA f32 16x16x4, accumulate over K ----
        v8f acc[MT][2];
#pragma unroll
        for (int mt = 0; mt < MT; ++mt) {
            v8f z = {0.f, 0.f, 0.f, 0.f, 0.f, 0.f, 0.f, 0.f};
            acc[mt][0] = z;
            acc[mt][1] = z;
        }
#pragma unroll
        for (int k = 0; k < KC; ++k) {
            const int kr = 4 * k + 2 * hi;  // this lane-half's K rows
            v2f b0, b1;                     // B frags for col tiles 0 and 1
            b0.x = yl[kr * C + lo];
            b0.y = yl[(kr + 1) * C + lo];
            b1.x = yl[kr * C + 16 + lo];
            b1.y = yl[(kr + 1) * C + 16 + lo];
#pragma unroll
            for (int mt = 0; mt < MT; ++mt) {
                v2f a;                      // A frag: S[mt*16+lo][kr..kr+1]
                a.x = s_S[(mt * 16 + lo) * D + kr];
                a.y = s_S[(mt * 16 + lo) * D + kr + 1];
                acc[mt][0] = __builtin_amdgcn_wmma_f32_16x16x4_f32(
                    false, a, false, b0, (short)0, acc[mt][0], false, false);
                acc[mt][1] = __builtin_amdgcn_wmma_f32_16x16x4_f32(
                    false, a, false, b1, (short)0, acc[mt][1], false, false);
            }
        }

        // ---- norm_c = sum_i y[i,c] * t[i,c] ----
        // D-tile layout: vgpr v, lane -> row mt*16 + v + 8*hi, col lo (+16 for tile 1)
        float p0 = 0.f, p1 = 0.f;
#pragma unroll
        for (int mt = 0; mt < MT; ++mt) {
#pragma unroll
            for (int v = 0; v < 8; ++v) {
                const int row = mt * 16 + v + 8 * hi;
                p0 += acc[mt][0][v] * yl[row * C + lo];
                p1 += acc[mt][1][v] * yl[row * C + 16 + lo];
            }
        }
        p0 += __shfl_xor(p0, 16, 32);  // combine the two row-halves
        p1 += __shfl_xor(p1, 16, 32);
        const float inv0 = 1.0f / (sqrtf(p0) + 1e-6f);
        const float inv1 = 1.0f / (sqrtf(p1) + 1e-6f);

        // ---- scale in LDS, then coalesced b128 stores to global ----
#pragma unroll
        for (int mt = 0; mt < MT; ++mt) {
#pragma unroll
            for (int v = 0; v < 8; ++v) {
                const int row = mt * 16 + v + 8 * hi;
                yl[row * C + lo]      = yl[row * C + lo] * inv0;
                yl[row * C + 16 + lo] = yl[row * C + 16 + lo] * inv1;
            }
        }
        asm volatile("s_wait_dscnt 0" ::: "memory");
#pragma unroll
        for (int i = lane; i < NV4; i += 32) oa4[i] = yl4[i];
    }
}

extern "C" void kernel_launch(void* const* d_in, const int* in_sizes, int n_in,
                              void* d_out, int out_size, void* d_ws, size_t ws_size,
                              hipStream_t stream) {
    // setup_inputs() order: x, S_H (16x16), S_O (48x48), idx_H, idx_O
    const float* x   = (const float*)d_in[0];
    const float* S_H = (const float*)d_in[1];
    const float* S_O = (const float*)d_in[2];
    float* out = (float*)d_out;

    const int nH = in_sizes[3] / 16;                 // 40000 atoms, d=16
    const int nO = in_sizes[4] / 48;                 // 20000 atoms, d=48
    const long long oBase = (long long)in_sizes[3];  // first O basis row

    const int blocksH = (nH + WAVES_PER_BLOCK - 1) / WAVES_PER_BLOCK;
    const int blocksO = (nO + WAVES_PER_BLOCK - 1) / WAVES_PER_BLOCK;

    l2norm_wmma_kernel<16><<<blocksH, 256, 0, stream>>>(x, S_H, out, nH, 0LL);
    l2norm_wmma_kernel<48><<<blocksO, 256, 0, stream>>>(x, S_O, out, nO, oBase);
}